// res_gcn_block_15616501088309
// MI455X (gfx1250) — compile-verified
//
#include <hip/hip_runtime.h>

typedef __attribute__((ext_vector_type(2))) float v2f;
typedef __attribute__((ext_vector_type(8))) float v8f;

#define B_  8
#define C_  256
#define N_  4096
#define K_  16

// ---------------------------------------------------------------------------
// Kernel 1: Y[b, 0:256, n]  = W1 @ relu(P[b])   (feats_center)
//           Y[b, 256:512,n] = W2 @ relu(P[b])   (gathered later for *_nn)
// One wave32 per 16x16 output tile, K-loop over 256 in steps of 4 using
// V_WMMA_F32_16X16X4_F32.  ~8.6 GFLOP total -> negligible vs write BW.
// ---------------------------------------------------------------------------
__global__ __launch_bounds__(32)
void res_gcn_gemm_wmma(const float* __restrict__ P,
                       const float* __restrict__ W1,
                       const float* __restrict__ W2,
                       float* __restrict__ Y)
{
    const int lane = threadIdx.x;      // 0..31
    const int mr   = lane & 15;        // M (for A) / N (for B) within tile
    const int hi   = lane >> 4;        // selects K pair (ISA 16x4 f32 layout)

    const int nt = blockIdx.x;         // 0..255  (N tiles)
    const int mt = blockIdx.y;         // 0..31   (M tiles over stacked [W1;W2])
    const int b  = blockIdx.z;         // 0..7
    const int n0 = nt * 16;
    const int m0 = mt * 16;

    // A operand: row m0+mr of stacked weight matrix [W1;W2] (512x256)
    const float* Wrow = (m0 < C_) ? (W1 + (size_t)(m0 + mr) * C_)
                                  : (W2 + (size_t)(m0 - C_ + mr) * C_);
    Wrow += 2 * hi;                                    // K = kk + 2*hi + {0,1}

    // B operand: relu(P[b, kk+2*hi+{0,1}, n0+mr]) — coalesced per 16-lane half
    const float* Pb   = P + (size_t)b * C_ * N_;
    const float* Pcol = Pb + (size_t)(2 * hi) * N_ + n0 + mr;

    v8f acc = {};
    #pragma unroll 4
    for (int kk = 0; kk < C_; kk += 4) {
        v2f a = *(const v2f*)(Wrow + kk);              // 8B load, adjacent K
        v2f bb;
        bb.x = fmaxf(Pcol[(size_t)kk * N_        ], 0.0f);   // fused relu
        bb.y = fmaxf(Pcol[(size_t)kk * N_ + N_   ], 0.0f);
        // D = A(16x4) * B(4x16) + C ; 8 args: neg_a,A,neg_b,B,c_mod,C,reuse_a,reuse_b
        acc = __builtin_amdgcn_wmma_f32_16x16x4_f32(
                  false, a, false, bb, (short)0, acc, false, false);
    }

    // C/D layout: VGPR v -> M = m0 + v + 8*hi ; N = n0 + mr
    float* yt = Y + ((size_t)b * 512 + m0 + 8 * hi) * N_ + n0 + mr;
    #pragma unroll
    for (int v = 0; v < 8; ++v)
        yt[(size_t)v * N_] = acc[v];
}

// ---------------------------------------------------------------------------
// Kernel 2: all four outputs via gathers of P / Y (L2-resident sources).
// Block = 256 threads handles one (b, n): lane t -> (c = ci*16 + t/16, k = t%16).
// Big streaming outputs (grouped, grouped_nn: 1 GB) use non-temporal stores so
// L2 stays dedicated to the gather sources.  K-reduction via 16-lane shfl_xor.
// ---------------------------------------------------------------------------
__global__ __launch_bounds__(256)
void res_gcn_scatter(const float* __restrict__ P,
                     const int*   __restrict__ IDX,
                     const float* __restrict__ Y,
                     float* __restrict__ out_feat,
                     float* __restrict__ out_center,
                     float* __restrict__ out_grp,
                     float* __restrict__ out_nn)
{
    const int n  = blockIdx.x;         // 0..4095
    const int b  = blockIdx.y;         // 0..7
    const int t  = threadIdx.x;        // 0..255
    const int k  = t & 15;
    const int cg = t >> 4;             // 0..15

    __shared__ int sidx[K_];
    if (t < K_) sidx[t] = IDX[((size_t)b * N_ + n) * K_ + t];
    __syncthreads();
    const int idx = sidx[k];

    const float* Pb = P + (size_t)b * C_ * N_;
    const float* Y1 = Y + (size_t)b * 512 * N_;        // W1 @ relu(P)
    const float* Y2 = Y1 + (size_t)C_ * N_;            // W2 @ relu(P)

    #pragma unroll
    for (int ci = 0; ci < 16; ++ci) {
        const int    c    = ci * 16 + cg;
        const size_t rowP = (size_t)c * N_;

        const float g  = fmaxf(Pb[rowP + idx], 0.0f);  // relu(points) gather
        const float nn = Y2[rowP + idx];               // (W2@feat) gather

        const size_t go = (((size_t)b * C_ + c) * N_ + n) * K_ + k;
        __builtin_nontemporal_store(g,  out_grp + go); // 64B/16-lane runs, NT
        __builtin_nontemporal_store(nn, out_nn  + go);

        // sum over k within each 16-lane group (butterfly)
        float s = nn;
        s += __shfl_xor(s, 1, 16);
        s += __shfl_xor(s, 2, 16);
        s += __shfl_xor(s, 4, 16);
        s += __shfl_xor(s, 8, 16);

        if (k == 0) {
            const size_t o = ((size_t)b * C_ + c) * N_ + n;
            const float  p = Pb[rowP + n];
            out_feat[o]   = (Y1[rowP + n] + s) * (1.0f / (K_ + 1)) + p;
            out_center[o] = fmaxf(p, 0.0f);
        }
    }
}

// ---------------------------------------------------------------------------
extern "C" void kernel_launch(void* const* d_in, const int* in_sizes, int n_in,
                              void* d_out, int out_size, void* d_ws, size_t ws_size,
                              hipStream_t stream)
{
    const float* P   = (const float*)d_in[0];   // (B, C, N) f32
    const float* W1  = (const float*)d_in[1];   // (C, C)
    const float* W2  = (const float*)d_in[2];   // (C, C)
    const int*   IDX = (const int*)  d_in[3];   // (B, N, K)

    float* Y = (float*)d_ws;                    // (B, 512, N) = 64 MB scratch

    float* out_feat   = (float*)d_out;                                  // (B,C,N)
    float* out_center = out_feat   + (size_t)B_ * C_ * N_;              // (B,C,N,1)
    float* out_grp    = out_center + (size_t)B_ * C_ * N_;              // (B,C,N,K)
    float* out_nn     = out_grp    + (size_t)B_ * C_ * N_ * K_;         // (B,C,N,K)

    dim3 g1(N_ / 16, 512 / 16, B_);             // 256 x 32 x 8 tiles
    res_gcn_gemm_wmma<<<g1, 32, 0, stream>>>(P, W1, W2, Y);

    dim3 g2(N_, B_, 1);                         // one block per (b, n)
    res_gcn_scatter<<<g2, 256, 0, stream>>>(P, IDX, Y, out_feat, out_center,
                                            out_grp, out_nn);
}